// bPC_SNNLayer_65274912965275
// MI455X (gfx1250) — compile-verified
//
#include <hip/hip_runtime.h>
#include <hip/hip_bf16.h>

// Elementwise SNN state update: 5x fp32[4096*4096] in, 5x fp32[4096*4096] out.
// Memory-bound: 640 MB / call -> ~27 us floor at 23.3 TB/s. Strategy:
//  - 128-bit vector loads/stores (global_load_b128 / global_store_b128)
//  - non-temporal hints (stream past 192MB L2; nothing is re-read)
//  - branchless cndmask/fma body, wave32-friendly 256-thread blocks
//  - exact-fit grid: one float4 per lane, no grid-stride loop.

typedef __attribute__((ext_vector_type(4))) float v4f;

// Folded config constants (from CFG in the reference):
//   dt=0.5, kappa_j=0.2, tau_j=8, gamma_m=0.1, R_m=5, tau_m=10,
//   thresh=1, t_ref=2, tau_tr=20
#define DT        0.5f
#define DT_TAU_J  0.0625f    // dt / tau_j
#define KAPPA_J   0.2f
#define DT_TAU_M  0.05f      // dt / tau_m
#define GAMMA_M   0.1f
#define R_M       5.0f
#define THRESH    1.0f
#define T_REF     2.0f
#define INV_TAU_TR 0.05f     // 1 / tau_tr

__global__ __launch_bounds__(256)
void snn_update_kernel(const float* __restrict__ gI,
                       const float* __restrict__ gV,
                       const float* __restrict__ gX,
                       const float* __restrict__ gJ,
                       const float* __restrict__ gR,
                       float* __restrict__ out,
                       unsigned int n_elems)   // total scalar elements (multiple of 4)
{
    const unsigned int n4 = n_elems >> 2;
    const unsigned int i  = blockIdx.x * blockDim.x + threadIdx.x;
    if (i >= n4) return;

    const v4f* __restrict__ Iv = (const v4f*)gI;
    const v4f* __restrict__ Vv = (const v4f*)gV;
    const v4f* __restrict__ Xv = (const v4f*)gX;
    const v4f* __restrict__ Jv = (const v4f*)gJ;
    const v4f* __restrict__ Rv = (const v4f*)gR;

    // Streaming loads: non-temporal (TH=NT) -> don't pollute L2 with
    // a 640MB one-shot stream.
    v4f I  = __builtin_nontemporal_load(Iv + i);
    v4f v0 = __builtin_nontemporal_load(Vv + i);
    v4f x0 = __builtin_nontemporal_load(Xv + i);
    v4f j0 = __builtin_nontemporal_load(Jv + i);
    v4f r0 = __builtin_nontemporal_load(Rv + i);

    v4f vo, so, xo, jo, ro;

#pragma unroll
    for (int c = 0; c < 4; ++c) {
        // refractory countdown
        float rc = fmaxf(r0[c] - DT, 0.0f);
        // synaptic current leaky integration: j += (dt/tau_j)*(-kappa*j + I)
        float jn = fmaf(DT_TAU_J, fmaf(-KAPPA_J, j0[c], I[c]), j0[c]);
        // membrane update, frozen while refractory
        float dv = DT_TAU_M * fmaf(-GAMMA_M, v0[c], R_M * jn);
        float vn = (rc > 0.0f) ? v0[c] : (v0[c] + dv);
        // threshold -> spike, reset, re-arm refractory
        bool  fired = vn > THRESH;
        float s  = fired ? 1.0f : 0.0f;
        float vr = fired ? 0.0f : vn;
        float rr = fired ? T_REF : rc;
        // low-pass spike trace: x = x*(1 - 1/tau_tr) + s
        float xn = fmaf(-INV_TAU_TR, x0[c], x0[c]) + s;

        vo[c] = vr; so[c] = s; xo[c] = xn; jo[c] = jn; ro[c] = rr;
    }

    // Output tuple (v, s, x, j, ref_count) concatenated flat.
    const unsigned int seg4 = n4;  // one tensor, in float4 units
    v4f* __restrict__ Ov = (v4f*)out;
    __builtin_nontemporal_store(vo, Ov + 0u * seg4 + i);
    __builtin_nontemporal_store(so, Ov + 1u * seg4 + i);
    __builtin_nontemporal_store(xo, Ov + 2u * seg4 + i);
    __builtin_nontemporal_store(jo, Ov + 3u * seg4 + i);
    __builtin_nontemporal_store(ro, Ov + 4u * seg4 + i);
}

extern "C" void kernel_launch(void* const* d_in, const int* in_sizes, int n_in,
                              void* d_out, int out_size, void* d_ws, size_t ws_size,
                              hipStream_t stream)
{
    const float* gI = (const float*)d_in[0];  // total_input_current
    const float* gV = (const float*)d_in[1];  // v
    const float* gX = (const float*)d_in[2];  // x
    const float* gJ = (const float*)d_in[3];  // j
    const float* gR = (const float*)d_in[4];  // ref_count
    float* out = (float*)d_out;

    const unsigned int n = (unsigned int)in_sizes[0];   // 4096*4096, multiple of 4
    const unsigned int n4 = n >> 2;
    const unsigned int threads = 256;                   // 8 wave32 per block
    const unsigned int blocks  = (n4 + threads - 1) / threads;

    snn_update_kernel<<<blocks, threads, 0, stream>>>(gI, gV, gX, gJ, gR, out, n);
}